// GCN_89086211653947
// MI455X (gfx1250) — compile-verified
//
#include <hip/hip_runtime.h>
#include <stdint.h>

#define N_NODES 100000
#define FEAT    128
#define HID     64
#define NCLASS  16
#define N_EDGES 1600000

#define USE_ASYNC_STAGE 1

typedef __attribute__((ext_vector_type(2))) float v2f;
typedef __attribute__((ext_vector_type(8))) float v8f;

// ---------------------------------------------------------------------------
// GEMM1: XW[N,64] = x[N,128] @ W1[128,64]   (f32 WMMA 16x16x4)
// W1 staged into LDS pre-swizzled as K-pairs: sB[k2*HID+n] = {W1[2k2][n], W1[2k2+1][n]}
// so each B operand is one aligned ds_load_b64 into an even VGPR pair.
// Staging uses GLOBAL_LOAD_ASYNC_TO_LDS_B32 (ASYNCcnt) doing the swizzle in-flight.
// 256 threads = 8 waves; each wave computes a 16-row x 64-col tile.
// ---------------------------------------------------------------------------
__global__ __launch_bounds__(256) void gcn_gemm1_wmma(
    const float* __restrict__ x, const float* __restrict__ W1,
    float* __restrict__ XW) {
  __shared__ v2f sB[(FEAT / 2) * HID];  // 32 KB
  const int tid = threadIdx.x;

#if USE_ASYNC_STAGE
  {
    const unsigned ldsBase = (unsigned)(uintptr_t)&sB[0];
    const unsigned long long gbase = (unsigned long long)(uintptr_t)W1;
#pragma unroll
    for (int i = 0; i < (FEAT * HID) / 256; ++i) {  // 32 elements per thread
      const int p = tid + i * 256;                  // n fastest -> coalesced
      const int k = p >> 6;
      const int n = p & 63;
      const unsigned gOff = (unsigned)p * 4u;
      const unsigned lOff =
          ldsBase + (unsigned)((((k >> 1) * HID + n) << 3) + ((k & 1) << 2));
      asm volatile("global_load_async_to_lds_b32 %0, %1, %2"
                   :: "v"(lOff), "v"(gOff), "s"(gbase) : "memory");
    }
    asm volatile("s_wait_asynccnt 0" ::: "memory");
  }
#else
  {
#pragma unroll
    for (int i = 0; i < (FEAT * HID) / 256; ++i) {
      const int p = tid + i * 256;
      const int k = p >> 6;
      const int n = p & 63;
      ((float*)&sB[(k >> 1) * HID + n])[k & 1] = W1[p];
    }
  }
#endif
  __syncthreads();

  const int wave = tid >> 5;
  const int lane = tid & 31;
  const int half = lane >> 4;   // 0: K pair {0,1}, 1: K pair {2,3}
  const int r    = lane & 15;   // row (A) / col (B) within tile
  const long m0  = ((long)blockIdx.x * 8 + wave) * 16;
  if (m0 >= N_NODES) return;    // uniform per-wave; EXEC stays all-ones

  v8f acc0 = {}, acc1 = {}, acc2 = {}, acc3 = {};
  const float* __restrict__ arow = x + (m0 + r) * FEAT;

#pragma unroll 4
  for (int k2 = 0; k2 < FEAT / 2; k2 += 2) {  // K=4 per iteration
    const int kk2 = k2 + half;
    const float2 av = *(const float2*)(arow + 2 * kk2);  // 8B aligned
    v2f a; a.x = av.x; a.y = av.y;
    const v2f* bp = sB + kk2 * HID + r;
    acc0 = __builtin_amdgcn_wmma_f32_16x16x4_f32(false, a, false, bp[0],  (short)0, acc0, false, false);
    acc1 = __builtin_amdgcn_wmma_f32_16x16x4_f32(false, a, false, bp[16], (short)0, acc1, false, false);
    acc2 = __builtin_amdgcn_wmma_f32_16x16x4_f32(false, a, false, bp[32], (short)0, acc2, false, false);
    acc3 = __builtin_amdgcn_wmma_f32_16x16x4_f32(false, a, false, bp[48], (short)0, acc3, false, false);
  }

  // C/D layout: VGPR i -> row (i + 8*half), col r
#pragma unroll
  for (int i = 0; i < 8; ++i) {
    float* o = XW + (m0 + i + 8 * half) * HID + r;
    o[0]  = acc0[i];
    o[16] = acc1[i];
    o[32] = acc2[i];
    o[48] = acc3[i];
  }
}

// ---------------------------------------------------------------------------
// GEMM2: HW[N,16] = H[N,64] @ W2[64,16]  (same scheme, plain swizzle stage)
// ---------------------------------------------------------------------------
__global__ __launch_bounds__(256) void gcn_gemm2_wmma(
    const float* __restrict__ H, const float* __restrict__ W2,
    float* __restrict__ HW) {
  __shared__ v2f sB[(HID / 2) * NCLASS];  // 4 KB
  const int tid = threadIdx.x;
#pragma unroll
  for (int i = 0; i < (HID * NCLASS) / 256; ++i) {  // 4 per thread
    const int p = tid + i * 256;
    const int k = p >> 4;
    const int n = p & 15;
    ((float*)&sB[(k >> 1) * NCLASS + n])[k & 1] = W2[p];
  }
  __syncthreads();

  const int wave = tid >> 5;
  const int lane = tid & 31;
  const int half = lane >> 4;
  const int r    = lane & 15;
  const long m0  = ((long)blockIdx.x * 8 + wave) * 16;
  if (m0 >= N_NODES) return;

  v8f acc = {};
  const float* __restrict__ arow = H + (m0 + r) * HID;
#pragma unroll 4
  for (int k2 = 0; k2 < HID / 2; k2 += 2) {
    const int kk2 = k2 + half;
    const float2 av = *(const float2*)(arow + 2 * kk2);
    v2f a; a.x = av.x; a.y = av.y;
    const v2f b = sB[kk2 * NCLASS + r];
    acc = __builtin_amdgcn_wmma_f32_16x16x4_f32(false, a, false, b, (short)0, acc, false, false);
  }
#pragma unroll
  for (int i = 0; i < 8; ++i)
    HW[(m0 + i + 8 * half) * NCLASS + r] = acc[i];
}

// ---------------------------------------------------------------------------
// SPMM d=64: 16 threads per edge, float4 per thread (coalesced gather)
// dst[row] += val * src[col]
// ---------------------------------------------------------------------------
__global__ __launch_bounds__(256) void gcn_spmm64(
    const int* __restrict__ rows, const int* __restrict__ cols,
    const float* __restrict__ vals, const float* __restrict__ src,
    float* __restrict__ dst) {
  const long gid = (long)blockIdx.x * 256 + threadIdx.x;
  const long e = gid >> 4;
  if (e >= N_EDGES) return;
  const int f0 = (int)(gid & 15) * 4;
  const int rw = rows[e];
  const int cl = cols[e];
  const float v = vals[e];
  const float4 s = *(const float4*)(src + (long)cl * HID + f0);
  float* d = dst + (long)rw * HID + f0;
  atomicAdd(d + 0, v * s.x);
  atomicAdd(d + 1, v * s.y);
  atomicAdd(d + 2, v * s.z);
  atomicAdd(d + 3, v * s.w);
}

// ---------------------------------------------------------------------------
// SPMM d=16: 4 threads per edge, float4 per thread
// ---------------------------------------------------------------------------
__global__ __launch_bounds__(256) void gcn_spmm16(
    const int* __restrict__ rows, const int* __restrict__ cols,
    const float* __restrict__ vals, const float* __restrict__ src,
    float* __restrict__ dst) {
  const long gid = (long)blockIdx.x * 256 + threadIdx.x;
  const long e = gid >> 2;
  if (e >= N_EDGES) return;
  const int f0 = (int)(gid & 3) * 4;
  const int rw = rows[e];
  const int cl = cols[e];
  const float v = vals[e];
  const float4 s = *(const float4*)(src + (long)cl * NCLASS + f0);
  float* d = dst + (long)rw * NCLASS + f0;
  atomicAdd(d + 0, v * s.x);
  atomicAdd(d + 1, v * s.y);
  atomicAdd(d + 2, v * s.z);
  atomicAdd(d + 3, v * s.w);
}

// ---------------------------------------------------------------------------
// Helpers: zero-fill, bias+relu, bias broadcast init
// ---------------------------------------------------------------------------
__global__ __launch_bounds__(256) void gcn_zero(float* __restrict__ p, long n) {
  const long i = (long)blockIdx.x * 256 + threadIdx.x;
  if (i < n) p[i] = 0.0f;
}

__global__ __launch_bounds__(256) void gcn_bias_relu64(
    float* __restrict__ h, const float* __restrict__ b1) {
  const long i = (long)blockIdx.x * 256 + threadIdx.x;
  if (i >= (long)N_NODES * HID) return;
  const float t = h[i] + b1[i & (HID - 1)];
  h[i] = t > 0.0f ? t : 0.0f;
}

__global__ __launch_bounds__(256) void gcn_init_out16(
    float* __restrict__ out, const float* __restrict__ b2) {
  const long i = (long)blockIdx.x * 256 + threadIdx.x;
  if (i >= (long)N_NODES * NCLASS) return;
  out[i] = b2[i & (NCLASS - 1)];
}

// ---------------------------------------------------------------------------
extern "C" void kernel_launch(void* const* d_in, const int* in_sizes, int n_in,
                              void* d_out, int out_size, void* d_ws, size_t ws_size,
                              hipStream_t stream) {
  const float* x    = (const float*)d_in[0];
  const float* vals = (const float*)d_in[1];
  const float* W1   = (const float*)d_in[2];
  const float* b1   = (const float*)d_in[3];
  const float* W2   = (const float*)d_in[4];
  const float* b2   = (const float*)d_in[5];
  const int*  rows  = (const int*)d_in[6];
  const int*  cols  = (const int*)d_in[7];
  float* out = (float*)d_out;

  float* XW = (float*)d_ws;                         // [N, 64]
  float* H  = XW + (size_t)N_NODES * HID;           // [N, 64]
  float* HW = H  + (size_t)N_NODES * HID;           // [N, 16]

  const int tiles = N_NODES / 16;                   // 6250 (exact)
  const int gemmBlocks = (tiles + 7) / 8;           // 782

  // 1) XW = x @ W1 (WMMA f32, async-LDS staged weights)
  gcn_gemm1_wmma<<<gemmBlocks, 256, 0, stream>>>(x, W1, XW);

  // 2) H = 0; H += A @ XW; H = relu(H + b1)
  {
    const long n = (long)N_NODES * HID;
    gcn_zero<<<(int)((n + 255) / 256), 256, 0, stream>>>(H, n);
  }
  gcn_spmm64<<<(int)(((long)N_EDGES * 16 + 255) / 256), 256, 0, stream>>>(
      rows, cols, vals, XW, H);
  gcn_bias_relu64<<<(int)(((long)N_NODES * HID + 255) / 256), 256, 0, stream>>>(H, b1);

  // 3) HW = H @ W2 (WMMA f32)
  gcn_gemm2_wmma<<<gemmBlocks, 256, 0, stream>>>(H, W2, HW);

  // 4) out = b2 (broadcast); out += A @ HW
  gcn_init_out16<<<(int)(((long)N_NODES * NCLASS + 255) / 256), 256, 0, stream>>>(out, b2);
  gcn_spmm16<<<(int)(((long)N_EDGES * 4 + 255) / 256), 256, 0, stream>>>(
      rows, cols, vals, HW, out);
}